// ChebNet_Structural_74577812128603
// MI455X (gfx1250) — compile-verified
//
#include <hip/hip_runtime.h>

// ---------------- problem constants (from reference) ----------------
#define N_NODES   100000
#define N_EDGES   1600000
#define N_GRAPHS  256
#define F_IN      128
#define C_CLOSE   16
#define KCHEB     5

typedef __attribute__((ext_vector_type(2))) float v2f;
typedef __attribute__((ext_vector_type(8))) float v8f;

// monotone float <-> uint ordering (for atomic segment-max)
__device__ __forceinline__ unsigned f2ord(float f) {
    unsigned u = __float_as_uint(f);
    return (u & 0x80000000u) ? ~u : (u | 0x80000000u);
}
__device__ __forceinline__ float ord2f(unsigned u) {
    return (u & 0x80000000u) ? __uint_as_float(u & 0x7fffffffu)
                             : __uint_as_float(~u);
}

// ---------------- graph prep ----------------
__global__ void k_degree(const int* __restrict__ src, float* __restrict__ deg, int nE) {
    int e = blockIdx.x * blockDim.x + threadIdx.x;
    if (e < nE) atomicAdd(&deg[src[e]], 1.0f);
}

// dis buffer holds deg on entry; transform in place, also compute w_diag
__global__ void k_dis_wdiag(const float* __restrict__ lmax, const int* __restrict__ batch,
                            float* __restrict__ dis, float* __restrict__ wdiag, int n) {
    int i = blockIdx.x * blockDim.x + threadIdx.x;
    if (i >= n) return;
    float d = dis[i];
    dis[i]   = (d > 0.0f) ? rsqrtf(fmaxf(d, 1.0f)) : 0.0f;
    wdiag[i] = 2.0f / lmax[batch[i]] - 1.0f;
}

__global__ void k_wedge(const int* __restrict__ src, const int* __restrict__ dst,
                        const int* __restrict__ batch, const float* __restrict__ lmax,
                        const float* __restrict__ dis, float* __restrict__ wedge, int nE) {
    int e = blockIdx.x * blockDim.x + threadIdx.x;
    if (e >= nE) return;
    int s = src[e], d = dst[e];
    wedge[e] = -2.0f * dis[s] * dis[d] / lmax[batch[s]];
}

// ---------------- Chebyshev recurrence pieces ----------------
// copy layer input X [n,F] into T-block 0 of Tcat [n, K*F]
__global__ void k_copyT0(const float* __restrict__ X, float* __restrict__ T0,
                         int F, int KF, int n) {
    int t = blockIdx.x * blockDim.x + threadIdx.x;
    if (t >= n * F) return;
    int row = t / F, f = t - row * F;
    T0[(size_t)row * KF + f] = X[(size_t)row * F + f];
}

// Tout = scale * wdiag * Tin  - (Tm2 ? Tm2 : 0)   (initializes Tout block)
__global__ void k_diag(const float* __restrict__ Tin, const float* __restrict__ Tm2,
                       float* __restrict__ Tout, const float* __restrict__ wdiag,
                       float scale, int F, int KF, int n) {
    int t = blockIdx.x * blockDim.x + threadIdx.x;
    if (t >= n * F) return;
    int row = t / F, f = t - row * F;
    size_t idx = (size_t)row * KF + f;
    float v = scale * wdiag[row] * Tin[idx];
    if (Tm2) v -= Tm2[idx];
    Tout[idx] = v;
}

// Tout[dst] += scale * wedge * Tin[src]    (4 features per thread, float4 gather)
__global__ void k_scatter(const float* __restrict__ Tin, float* __restrict__ Tout,
                          const int* __restrict__ src, const int* __restrict__ dst,
                          const float* __restrict__ wedge, float scale,
                          int Fv /* F/4 */, int KF, int total) {
    int t = blockIdx.x * blockDim.x + threadIdx.x;
    if (t >= total) return;
    int e  = t / Fv;
    int f0 = (t - e * Fv) * 4;
    float w = scale * wedge[e];
    size_t si = (size_t)src[e] * KF + f0;
    size_t di = (size_t)dst[e] * KF + f0;
    float4 xv = *(const float4*)(Tin + si);
    atomicAdd(&Tout[di + 0], w * xv.x);
    atomicAdd(&Tout[di + 1], w * xv.y);
    atomicAdd(&Tout[di + 2], w * xv.z);
    atomicAdd(&Tout[di + 3], w * xv.w);
}

// ---------------- fused Chebyshev GEMM:  H = relu(Tcat[N,KF] @ W[KF,FO] + b) ----------
// One wave -> 16 rows x FO cols (FO/16 16x16 tiles) via V_WMMA_F32_16X16X4_F32, K-loop in steps of 4.
// Whole weight matrix (<= 80 KB) staged in LDS (320 KB per WGP on CDNA5).
template <int KF, int FO>
__global__ void k_cheb_gemm(const float* __restrict__ T, const float* __restrict__ W,
                            const float* __restrict__ bias, float* __restrict__ H,
                            int n_rowgroups) {
    __shared__ float lw[KF * FO];
    for (int i = threadIdx.x; i < KF * FO; i += blockDim.x) lw[i] = W[i];
    __syncthreads();

    const int wave = threadIdx.x >> 5;
    const int lane = threadIdx.x & 31;
    const int rg   = blockIdx.x * 8 + wave;     // 8 waves per block, 16 rows each
    if (rg >= n_rowgroups) return;

    const int row0  = rg * 16;
    const int m     = lane & 15;                // A: M index / B,C: N index
    const int khalf = (lane >> 4) * 2;          // 0 or 2 (K half per ISA A/B layout)
    const float* arow = T + (size_t)(row0 + m) * KF + khalf;

    for (int ct = 0; ct < FO / 16; ++ct) {
        const int col0 = ct * 16;
        v8f acc = {};
        for (int kk = 0; kk < KF; kk += 4) {
            v2f a = *(const v2f*)(arow + kk);   // A[m][kk+khalf], A[m][kk+khalf+1]
            v2f b;
            b.x = lw[(kk + khalf) * FO + col0 + m];       // B[kk+khalf][col]
            b.y = lw[(kk + khalf + 1) * FO + col0 + m];   // B[kk+khalf+1][col]
            acc = __builtin_amdgcn_wmma_f32_16x16x4_f32(
                false, a, false, b, (short)0, acc, false, false);
        }
        // C/D layout: lanes 0-15 -> M = v, lanes 16-31 -> M = v+8; N = lane&15
        const int col  = col0 + m;
        const float bb = bias[col];
        const int rbase = row0 + ((lane >> 4) * 8);
        #pragma unroll
        for (int v = 0; v < 8; ++v) {
            float val = acc[v] + bb;
            H[(size_t)(rbase + v) * FO + col] = val > 0.0f ? val : 0.0f;
        }
    }
}

// ---------------- pooling + head ----------------
__global__ void k_score(const float* __restrict__ close, const float* __restrict__ cw,
                        const float* __restrict__ cb, float* __restrict__ s, int n) {
    int i = blockIdx.x * blockDim.x + threadIdx.x;
    if (i >= n) return;
    float acc = cb[0];
    #pragma unroll
    for (int j = 0; j < C_CLOSE; ++j) acc += close[(size_t)i * C_CLOSE + j] * cw[j];
    s[i] = acc;
}

__global__ void k_fill_ord_neginf(unsigned* __restrict__ p, int n) {
    int i = blockIdx.x * blockDim.x + threadIdx.x;
    if (i < n) p[i] = 0x007FFFFFu;   // f2ord(-inf)
}

__global__ void k_smax(const float* __restrict__ s, const int* __restrict__ batch,
                       unsigned* __restrict__ smax, int n) {
    int i = blockIdx.x * blockDim.x + threadIdx.x;
    if (i >= n) return;
    atomicMax(&smax[batch[i]], f2ord(s[i]));
}

__global__ void k_expsum(float* __restrict__ s, const int* __restrict__ batch,
                         const unsigned* __restrict__ smax, float* __restrict__ den,
                         float* __restrict__ cnt, int n) {
    int i = blockIdx.x * blockDim.x + threadIdx.x;
    if (i >= n) return;
    int b = batch[i];
    float e = expf(s[i] - ord2f(smax[b]));
    s[i] = e;                       // reuse score buffer for exp values
    atomicAdd(&den[b], e);
    atomicAdd(&cnt[b], 1.0f);
}

__global__ void k_wmax(const float* __restrict__ e, const int* __restrict__ batch,
                       const float* __restrict__ den, const float* __restrict__ cnt,
                       const float* __restrict__ h, unsigned* __restrict__ g, int total) {
    int t = blockIdx.x * blockDim.x + threadIdx.x;
    if (t >= total) return;
    int nrow = t >> 6, f = t & 63;
    int b = batch[nrow];
    float w = e[nrow] / den[b] * cnt[b];
    atomicMax(&g[(size_t)b * 64 + f], f2ord(w * h[(size_t)nrow * 64 + f]));
}

__global__ void k_head(const unsigned* __restrict__ g, const float* __restrict__ a1w,
                       const float* __restrict__ a1b, const float* __restrict__ a2w,
                       const float* __restrict__ a2b, float* __restrict__ out) {
    int b = threadIdx.x;            // one thread per graph, 256 threads
    float gr[64];
    #pragma unroll
    for (int j = 0; j < 64; ++j) gr[j] = ord2f(g[(size_t)b * 64 + j]);
    float o = a2b[0];
    #pragma unroll
    for (int i = 0; i < 16; ++i) {
        float acc = a1b[i];
        for (int j = 0; j < 64; ++j) acc += gr[j] * a1w[j * 16 + i];
        acc = acc > 0.0f ? acc : 0.0f;
        o += acc * a2w[i];
    }
    out[b] = o;
}

// ---------------- host orchestration ----------------
static inline int gridFor(long long total, int block) {
    return (int)((total + block - 1) / block);
}

extern "C" void kernel_launch(void* const* d_in, const int* in_sizes, int n_in,
                              void* d_out, int out_size, void* d_ws, size_t ws_size,
                              hipStream_t stream) {
    (void)in_sizes; (void)n_in; (void)out_size; (void)ws_size;
    const float* x         = (const float*)d_in[0];
    const float* closeness = (const float*)d_in[1];
    const float* lmax      = (const float*)d_in[2];
    const int*   src       = (const int*)d_in[3];
    const int*   dst       = (const int*)d_in[4];
    const int*   batch     = (const int*)d_in[5];
    const float* w1 = (const float*)d_in[6];  const float* b1 = (const float*)d_in[7];
    const float* w2 = (const float*)d_in[8];  const float* b2 = (const float*)d_in[9];
    const float* w3 = (const float*)d_in[10]; const float* b3 = (const float*)d_in[11];
    const float* cw = (const float*)d_in[12]; const float* cb = (const float*)d_in[13];
    const float* a1w = (const float*)d_in[14]; const float* a1b = (const float*)d_in[15];
    const float* a2w = (const float*)d_in[16]; const float* a2b = (const float*)d_in[17];
    float* out = (float*)d_out;

    // ---- carve workspace ----
    char* ws = (char*)d_ws;
    size_t off = 0;
    auto carve = [&](size_t bytes) -> char* {
        char* p = ws + off;
        off = (off + bytes + 255) & ~(size_t)255;
        return p;
    };
    float*    dis   = (float*)carve((size_t)N_NODES * 4);       // deg, then dis (in place)
    float*    wdiag = (float*)carve((size_t)N_NODES * 4);
    float*    wedge = (float*)carve((size_t)N_EDGES * 4);
    float*    Tcat  = (float*)carve((size_t)N_NODES * (KCHEB * F_IN) * 4); // 256 MB, reused per layer
    float*    h1    = (float*)carve((size_t)N_NODES * 32 * 4);
    float*    h2    = (float*)carve((size_t)N_NODES * 64 * 4);
    float*    h3    = (float*)carve((size_t)N_NODES * 64 * 4);
    float*    sbuf  = (float*)carve((size_t)N_NODES * 4);
    unsigned* smax  = (unsigned*)carve((size_t)N_GRAPHS * 4);
    float*    den   = (float*)carve((size_t)N_GRAPHS * 4);
    float*    cnt   = (float*)carve((size_t)N_GRAPHS * 4);
    unsigned* gbuf  = (unsigned*)carve((size_t)N_GRAPHS * 64 * 4);

    const int BLK = 256;

    // ---- graph prep ----
    hipMemsetAsync(dis, 0, (size_t)N_NODES * 4, stream);
    k_degree<<<gridFor(N_EDGES, BLK), BLK, 0, stream>>>(src, dis, N_EDGES);
    k_dis_wdiag<<<gridFor(N_NODES, BLK), BLK, 0, stream>>>(lmax, batch, dis, wdiag, N_NODES);
    k_wedge<<<gridFor(N_EDGES, BLK), BLK, 0, stream>>>(src, dst, batch, lmax, dis, wedge, N_EDGES);

    const int n_rg = N_NODES / 16;              // 6250 row-groups of 16
    const int gemm_blocks = (n_rg + 7) / 8;

    // ---- one ChebConv layer ----
    auto cheb_layer = [&](const float* X, int F, const float* W, const float* bias,
                          float* H, int FO) {
        const int KF = KCHEB * F;
        const int Fv = F / 4;
        const int sc_total = N_EDGES * Fv;
        // T0 = X
        k_copyT0<<<gridFor((long long)N_NODES * F, BLK), BLK, 0, stream>>>(X, Tcat, F, KF, N_NODES);
        // T1 = L T0
        k_diag<<<gridFor((long long)N_NODES * F, BLK), BLK, 0, stream>>>(
            Tcat, (const float*)nullptr, Tcat + F, wdiag, 1.0f, F, KF, N_NODES);
        k_scatter<<<gridFor(sc_total, BLK), BLK, 0, stream>>>(
            Tcat, Tcat + F, src, dst, wedge, 1.0f, Fv, KF, sc_total);
        // T_k = 2 L T_{k-1} - T_{k-2}
        for (int k = 2; k < KCHEB; ++k) {
            k_diag<<<gridFor((long long)N_NODES * F, BLK), BLK, 0, stream>>>(
                Tcat + (size_t)(k - 1) * F, Tcat + (size_t)(k - 2) * F,
                Tcat + (size_t)k * F, wdiag, 2.0f, F, KF, N_NODES);
            k_scatter<<<gridFor(sc_total, BLK), BLK, 0, stream>>>(
                Tcat + (size_t)(k - 1) * F, Tcat + (size_t)k * F,
                src, dst, wedge, 2.0f, Fv, KF, sc_total);
        }
        // H = relu(Tcat @ Wflat + b)  (Wflat == stacked W[k], exactly row-major [KF,FO])
        if (F == 128)      k_cheb_gemm<KCHEB * 128, 32><<<gemm_blocks, 256, 0, stream>>>(Tcat, W, bias, H, n_rg);
        else if (F == 32)  k_cheb_gemm<KCHEB * 32,  64><<<gemm_blocks, 256, 0, stream>>>(Tcat, W, bias, H, n_rg);
        else               k_cheb_gemm<KCHEB * 64,  64><<<gemm_blocks, 256, 0, stream>>>(Tcat, W, bias, H, n_rg);
        (void)FO;
    };

    cheb_layer(x,  128, w1, b1, h1, 32);
    cheb_layer(h1, 32,  w2, b2, h2, 64);
    cheb_layer(h2, 64,  w3, b3, h3, 64);

    // ---- pooling ----
    k_score<<<gridFor(N_NODES, BLK), BLK, 0, stream>>>(closeness, cw, cb, sbuf, N_NODES);
    k_fill_ord_neginf<<<gridFor(N_GRAPHS, BLK), BLK, 0, stream>>>(smax, N_GRAPHS);
    k_fill_ord_neginf<<<gridFor(N_GRAPHS * 64, BLK), BLK, 0, stream>>>(gbuf, N_GRAPHS * 64);
    hipMemsetAsync(den, 0, (size_t)N_GRAPHS * 4, stream);
    hipMemsetAsync(cnt, 0, (size_t)N_GRAPHS * 4, stream);
    k_smax<<<gridFor(N_NODES, BLK), BLK, 0, stream>>>(sbuf, batch, smax, N_NODES);
    k_expsum<<<gridFor(N_NODES, BLK), BLK, 0, stream>>>(sbuf, batch, smax, den, cnt, N_NODES);
    k_wmax<<<gridFor((long long)N_NODES * 64, BLK), BLK, 0, stream>>>(
        sbuf, batch, den, cnt, h3, gbuf, N_NODES * 64);

    // ---- head ----
    k_head<<<1, N_GRAPHS, 0, stream>>>(gbuf, a1w, a1b, a2w, a2b, out);
}